// OneBodyReduction_7043746365490
// MI455X (gfx1250) — compile-verified
//
#include <hip/hip_runtime.h>

// Problem constants (from reference)
#define RR        256      // R*R
#define CCH       16       // channels
#define NOUT      32       // outputs
#define FEAT_PER_ATOM 4096 // 256*16 floats
#define OUT_PER_ATOM  512  // 32*16 floats
#define W_PER_Z   8192     // 256*32 floats

#define WAVES_PER_WG   4
#define NUM_WGS        512
#define ATOMS_PER_WAVE 4   // 512 * 4 * 4 = 8192 atoms

#define W_LDS_FLOATS (5 * W_PER_Z)                               // 160 KB
#define SMEM_FLOATS  (W_LDS_FLOATS + WAVES_PER_WG * 2 * FEAT_PER_ATOM) // 288 KB

typedef __attribute__((ext_vector_type(2))) float v2f;
typedef __attribute__((ext_vector_type(4))) float v4f;
typedef __attribute__((ext_vector_type(8))) float v8f;
typedef __attribute__((ext_vector_type(4))) int   v4i;

typedef __attribute__((address_space(1))) v4i* g_v4i_ptr;  // global
typedef __attribute__((address_space(3))) v4i* l_v4i_ptr;  // LDS

__device__ __forceinline__ unsigned lds_off(const void* p) {
  // generic LDS address: low 32 bits are the LDS byte offset (ISA 10.2 aperture map)
  return (unsigned)(unsigned long long)p;
}

__device__ __forceinline__ void async_copy_b128(const float* gsrc, float* ldst) {
#if __has_builtin(__builtin_amdgcn_global_load_async_to_lds_b128)
  __builtin_amdgcn_global_load_async_to_lds_b128(
      (g_v4i_ptr)(float*)gsrc, (l_v4i_ptr)ldst, /*offset=*/0, /*cpol=*/0);
#else
  asm volatile("global_load_async_to_lds_b128 %0, %1, off"
               :: "v"(lds_off(ldst)), "v"(gsrc)
               : "memory");
#endif
}

template <int N>
__device__ __forceinline__ void wait_asynccnt() {
#if __has_builtin(__builtin_amdgcn_s_wait_asynccnt)
  __builtin_amdgcn_s_wait_asynccnt(N);
#else
  asm volatile("s_wait_asynccnt %0" :: "i"(N));
#endif
}

__global__ __launch_bounds__(WAVES_PER_WG * 32, 1)
void onebody_reduction_kernel(const int*   __restrict__ zarr,
                              const float* __restrict__ feat,
                              const float* __restrict__ coeff,
                              float*       __restrict__ out) {
  extern __shared__ float smem[];
  float* Wl = smem;                       // [5][256][32]

  const int tid  = threadIdx.x;
  const int lane = tid & 31;
  const int wave = tid >> 5;
  const int h    = lane >> 4;             // lane half (K pairing in fp32 WMMA layout)
  const int c16  = lane & 15;

  // ---- cooperative async preload of the 5 used coefficient matrices into LDS
  {
    const int zr[5] = {1, 6, 7, 8, 9};
    #pragma unroll
    for (int s = 0; s < 5; ++s) {
      const float* src = coeff + (long)zr[s] * W_PER_Z;
      float*       dst = Wl + s * W_PER_Z;
      #pragma unroll
      for (int i = 0; i < 16; ++i) {
        const int off = (i * 128 + tid) * 4;     // 2048 float4 per matrix
        async_copy_b128(src + off, dst + off);
      }
    }
  }

  float* fb0 = smem + W_LDS_FLOATS + (wave * 2 + 0) * FEAT_PER_ATOM;
  float* fb1 = smem + W_LDS_FLOATS + (wave * 2 + 1) * FEAT_PER_ATOM;

  const int wave_id = blockIdx.x * WAVES_PER_WG + wave;
  const int n0      = wave_id * ATOMS_PER_WAVE;

  // prefetch first atom's features into buffer 0
  {
    const float* src = feat + (long)n0 * FEAT_PER_ATOM;
    #pragma unroll
    for (int i = 0; i < 32; ++i) {
      const int off = (i * 32 + lane) * 4;       // 1024 float4 per atom
      async_copy_b128(src + off, fb0 + off);
    }
  }
  wait_asynccnt<0>();   // W table + first feat tile resident
  __syncthreads();      // W table visible to all waves

  for (int a = 0; a < ATOMS_PER_WAVE; ++a) {
    const int n = n0 + a;
    float* cur = (a & 1) ? fb1 : fb0;
    float* nxt = (a & 1) ? fb0 : fb1;

    if (a + 1 < ATOMS_PER_WAVE) {
      const float* src = feat + (long)(n + 1) * FEAT_PER_ATOM;
      #pragma unroll
      for (int i = 0; i < 32; ++i) {
        const int off = (i * 32 + lane) * 4;
        async_copy_b128(src + off, nxt + off);
      }
      // async loads complete in order: <=32 outstanding => current buffer done
      wait_asynccnt<32>();
    } else {
      wait_asynccnt<0>();
    }

    const int z    = zarr[n];
    const int slot = (z == 6) ? 1 : (z == 7) ? 2 : (z == 8) ? 3 : (z == 9) ? 4 : 0;

    // D[M=c][N=o] = sum_r feat^T[c][r] * W[r][o]   (two 16-wide o tiles)
    v8f acc0 = {};
    v8f acc1 = {};
    int aidx = (2 * h) * CCH + c16;                    // feat LDS index, +64/step
    int bidx = slot * W_PER_Z + (2 * h) * NOUT + c16;  // W LDS index,   +128/step
    #pragma unroll 4
    for (int kk = 0; kk < 64; ++kk) {                  // K = 4 per step
      v2f A;  A.x  = cur[aidx];      A.y  = cur[aidx + 16];
      v2f B0; B0.x = Wl[bidx];       B0.y = Wl[bidx + 32];
      v2f B1; B1.x = Wl[bidx + 16];  B1.y = Wl[bidx + 48];
      acc0 = __builtin_amdgcn_wmma_f32_16x16x4_f32(false, A, false, B0,
                                                   (short)0, acc0, false, false);
      acc1 = __builtin_amdgcn_wmma_f32_16x16x4_f32(false, A, false, B1,
                                                   (short)0, acc1, false, false);
      aidx += 4 * CCH;
      bidx += 4 * NOUT;
    }

    // lane l, VGPR j of acc_t holds out[n][o = c16 + 16*t][c = 8*h + j]
    float* ob = out + (long)n * OUT_PER_ATOM + 8 * h;
    *(v4f*)(ob + c16 * CCH)            = (v4f){acc0[0], acc0[1], acc0[2], acc0[3]};
    *(v4f*)(ob + c16 * CCH + 4)        = (v4f){acc0[4], acc0[5], acc0[6], acc0[7]};
    *(v4f*)(ob + (c16 + 16) * CCH)     = (v4f){acc1[0], acc1[1], acc1[2], acc1[3]};
    *(v4f*)(ob + (c16 + 16) * CCH + 4) = (v4f){acc1[4], acc1[5], acc1[6], acc1[7]};
  }
}

extern "C" void kernel_launch(void* const* d_in, const int* in_sizes, int n_in,
                              void* d_out, int out_size, void* d_ws, size_t ws_size,
                              hipStream_t stream) {
  const int*   z     = (const int*)  d_in[0];   // atomic_numbers
  const float* feat  = (const float*)d_in[1];   // feat_ten
  const float* coeff = (const float*)d_in[2];   // reduction_coefficients
  float*       out   = (float*)d_out;

  const size_t shmem = (size_t)SMEM_FLOATS * sizeof(float);   // 288 KB (<= 320 KB/WGP)
  (void)hipFuncSetAttribute((const void*)onebody_reduction_kernel,
                            hipFuncAttributeMaxDynamicSharedMemorySize, (int)shmem);
  hipLaunchKernelGGL(onebody_reduction_kernel,
                     dim3(NUM_WGS), dim3(WAVES_PER_WG * 32), shmem, stream,
                     z, feat, coeff, out);
}